// GaussianVector_box_16020228014570
// MI455X (gfx1250) — compile-verified
//
#include <hip/hip_runtime.h>
#include <math.h>

typedef float v2f __attribute__((ext_vector_type(2)));
typedef float v8f __attribute__((ext_vector_type(8)));

#define OUT_W 1024
#define OUT_H 1024
#define BN    32768      // B*N = 128*256
#define ROWS  65536      // 2*B*N  (x-rows then y-rows)
#define RGRPS 4096       // ROWS / 16
#define LOG2E 1.4426950408889634f

// One wave handles 16 output rows x 256 columns (one column-quarter).
// Exponent of the Gaussian evaluated as a quadratic in p via
// V_WMMA_F32_16X16X4_F32:  E[16x16] = A[16x4(a,b,d,0)] * B[4x16(p^2,p,1,0)]
__global__ __launch_bounds__(256)
void gaussian_vec_wmma(const float* __restrict__ centers,
                       const float* __restrict__ whs,
                       float* __restrict__ out)
{
    const int lane   = threadIdx.x & 31;
    const int half   = lane & 15;
    const int upper  = lane >> 4;                 // 0: lanes 0-15, 1: lanes 16-31
    const int waveId = threadIdx.x >> 5;
    const int gWave  = blockIdx.x * 8 + waveId;   // 0 .. 16383
    const int rg     = gWave >> 2;                // row group 0 .. 4095
    const int quart  = gWave & 3;                 // column quarter 0 .. 3

    const int rowBase = rg << 4;                  // first of 16 rows
    const int myRow   = rowBase + half;           // row this lane parameterizes
    const bool isY    = myRow >= BN;
    const int boxIdx  = isY ? (myRow - BN) : myRow;

    const float cx = centers[2 * boxIdx + 0];
    const float cy = centers[2 * boxIdx + 1];
    const float wx = whs[2 * boxIdx + 0];
    const float wy = whs[2 * boxIdx + 1];

    // sigma chain with floor-divs, exactly as the reference
    const float ksw = floorf(wx * 0.5f) * 2.0f - 1.0f;
    const float rw  = floorf((ksw - 1.0f) * 0.5f);
    const float sw  = floorf(rw / 3.0f);
    const float ksh = floorf(wy * 0.5f) * 2.0f - 1.0f;
    const float rh  = floorf((ksh - 1.0f) * 0.5f);
    const float sh  = floorf(rh / 3.0f);

    const bool zero_box = ((cx + cy) + (wx + wy)) == 0.0f;

    const float x   = truncf(cx);                 // SCALE == 1
    const float y   = truncf(cy);
    const float ul0 = x - rw, ul1 = y - rh;
    const float br0 = x + rw + 1.0f, br1 = y + rh + 1.0f;

    const bool in0 = (ul0 >= 0.0f) & (ul0 <= (float)OUT_W) &
                     (ul1 >= 0.0f) & (ul1 <= (float)OUT_H);
    const bool in1 = (br0 >= 0.0f) & (br0 <= (float)OUT_W) &
                     (br1 >= 0.0f) & (br1 <= (float)OUT_H);
    const bool active = (!zero_box) & (sw != 0.0f) & (sh != 0.0f) & (in0 | in1);

    // axis-specific parameters for this row
    const float c = isY ? y : x;
    const float s = isY ? sh : sw;
    float lo = isY ? ul1 : ul0;
    float hi = isY ? br1 : br0;
    if (!active) { lo = 1.0e30f; hi = -1.0e30f; } // empty interval -> all zeros

    const float ss  = (s == 0.0f) ? 1.0f : s;     // "safe" sigma (row inactive anyway)
    const float inv = 0.5f / (ss * ss);
    // exponent(p) = -(p-c)^2 * inv; fold log2(e) so exp() becomes a bare v_exp_f32
    const float ka = -inv * LOG2E;
    const float kb = (2.0f * c * inv) * LOG2E;
    const float kd = (-c * c * inv) * LOG2E;

    // A matrix (16x4): lanes 0-15 -> {K0=a, K1=b}; lanes 16-31 -> {K2=d, K3=0}
    v2f A;
    A.x = upper ? kd : ka;
    A.y = upper ? 0.0f : kb;

    // Per-D-VGPR row bounds: D VGPR v maps to row (v + 8*upper); lane r holds row r's params
    float lov[8], hiv[8];
#pragma unroll
    for (int v = 0; v < 8; ++v) {
        lov[v] = __shfl(lo, v + 8 * upper, 32);
        hiv[v] = __shfl(hi, v + 8 * upper, 32);
    }

    // D layout: lane owns column (lane&15); VGPR v -> row (v + 8*upper)
    const size_t outBase = (size_t)(rowBase + 8 * upper) * OUT_W + (size_t)half;

    const int t0 = quart * 16;
#pragma unroll 4
    for (int t = t0; t < t0 + 16; ++t) {
        const float p = (float)(t * 16 + half);   // this lane's column position

        // B matrix (4x16): lanes 0-15 -> {K0=p^2, K1=p}; lanes 16-31 -> {K2=1, K3=0}
        v2f Bv;
        Bv.x = upper ? 1.0f : p * p;
        Bv.y = upper ? 0.0f : p;

        v8f acc = {};
        acc = __builtin_amdgcn_wmma_f32_16x16x4_f32(
            /*neg_a=*/false, A, /*neg_b=*/false, Bv,
            /*c_mod=*/(short)0, acc, /*reuse_a=*/false, /*reuse_b=*/false);

#pragma unroll
        for (int v = 0; v < 8; ++v) {
            const float e = acc[v];               // log2-domain exponent
#if __has_builtin(__builtin_amdgcn_exp2f)
            const float g = __builtin_amdgcn_exp2f(e);
#else
            const float g = exp2f(e);
#endif
            const float val = (p >= lov[v]) & (p < hiv[v]) ? g : 0.0f;
            __builtin_nontemporal_store(
                val, out + outBase + (size_t)v * OUT_W + (size_t)(t * 16));
        }
    }
}

extern "C" void kernel_launch(void* const* d_in, const int* in_sizes, int n_in,
                              void* d_out, int out_size, void* d_ws, size_t ws_size,
                              hipStream_t stream) {
    const float* centers = (const float*)d_in[0];  // boxes_center [128,256,2] f32
    const float* whs     = (const float*)d_in[1];  // boxes_wh     [128,256,2] f32
    float* out = (float*)d_out;                    // [vector_x | vector_y] f32

    // 16384 waves: 4096 row-groups x 4 column-quarters; 8 waves per 256-thread block
    dim3 grid(2048), block(256);
    hipLaunchKernelGGL(gaussian_vec_wmma, grid, block, 0, stream, centers, whs, out);
}